// SimpleVQ_23785528885835
// MI455X (gfx1250) — compile-verified
//
#include <hip/hip_runtime.h>

typedef float v2f __attribute__((ext_vector_type(2)));
typedef float v8f __attribute__((ext_vector_type(8)));

#define DDIM 64
#define LCODES 512
#define CBS 68              // padded LDS row stride (floats): bank-conflict-free for frag reads
#define CHUNK 64            // codes per LDS chunk
#define NCHUNKS (LCODES / CHUNK)
#define WAVES 8
#define TILE 16             // tokens per wave

// ---------------------------------------------------------------------------
// Kernel 1: build codebook (RMS-normalized interleaved sinusoid PE, *64^-0.25)
// ---------------------------------------------------------------------------
__global__ __launch_bounds__(64) void vq_codebook(float* __restrict__ cb,
                                                  float* __restrict__ c2) {
    __shared__ float sh[64];
    const int l = blockIdx.x;
    const int d = threadIdx.x;
    const float ex   = -(2.0f * (float)(d >> 1)) / 64.0f;
    const float freq = powf(10000.0f, ex);
    const float ang  = (float)l * freq;
    const float val  = (d & 1) ? cosf(ang) : sinf(ang);
    sh[d] = val * val;
    __syncthreads();
    for (int s = 32; s > 0; s >>= 1) {
        if (d < s) sh[d] += sh[d + s];
        __syncthreads();
    }
    const float sum = sh[0];
    const float rms = sqrtf(sum * (1.0f / 64.0f) + 1e-6f);
    const float sc  = 0.35355339059327373f;   // 64^-0.25
    cb[l * DDIM + d] = sc * val / rms;
    if (d == 0) c2[l] = (sc * sc) * sum / (rms * rms);
}

// ---------------------------------------------------------------------------
// Kernel 2: main VQ — WMMA f32 16x16x4 cross-term GEMM + min/argmin + gather
// ---------------------------------------------------------------------------
__global__ __launch_bounds__(256) void vq_main(const float* __restrict__ vecs,
                                               const float* __restrict__ mask,
                                               const float* __restrict__ cb,
                                               const float* __restrict__ c2,
                                               float* __restrict__ out_hat,
                                               float* __restrict__ out_z,
                                               float* __restrict__ out_e,
                                               float* __restrict__ partials,
                                               int H, int S) {
    __shared__ float lds_tok[WAVES * TILE * CBS];   // 34816 B
    __shared__ float lds_cb[CHUNK * CBS];           // 17408 B
    __shared__ float lds_part[WAVES];

    const int tid  = threadIdx.x;
    const int wave = tid >> 5;
    const int lane = tid & 31;
    const int half = lane >> 4;     // 0 or 1
    const int lx   = lane & 15;

    const long tile_id = (long)blockIdx.x * WAVES + wave;
    const long tok0    = tile_id * TILE;

    // ---- stage this wave's 16x64 token tile into padded LDS (coalesced b64)
    float* tk = &lds_tok[wave * TILE * CBS];
    {
        const float* src = vecs + tok0 * DDIM;
        #pragma unroll
        for (int i = 0; i < 16; ++i) {
            const int e  = (i * 32 + lane) * 2;      // 1024 floats total
            const v2f d2 = *(const v2f*)(src + e);
            const int t  = e / DDIM, dd = e % DDIM;
            *(v2f*)(tk + t * CBS + dd) = d2;
        }
    }

    // ---- A fragments for WMMA f32 16x16x4: frag j covers K = 4j..4j+3
    //      lanes 0-15: (K=4j,4j+1), lanes 16-31: (K=4j+2,4j+3), M = lane%16
    v2f a[16];
    #pragma unroll
    for (int j = 0; j < 16; ++j)
        a[j] = *(const v2f*)(tk + lx * CBS + 4 * j + 2 * half);

    // ---- |v|^2 per token (lane holds token lx after xor-16 combine)
    float v2p = 0.0f;
    #pragma unroll
    for (int j = 0; j < 16; ++j) v2p += a[j].x * a[j].x + a[j].y * a[j].y;
    v2p += __shfl_xor(v2p, 16);
    float v2m[8];
    #pragma unroll
    for (int r = 0; r < 8; ++r) v2m[r] = __shfl(v2p, r + 8 * half);

    // ---- running min of (|c|^2 - 2 v.c); |v|^2 added at the end
    float best[8];
    int   bidx[8];
    #pragma unroll
    for (int r = 0; r < 8; ++r) { best[r] = 3.4e38f; bidx[r] = 0; }

    for (int ch = 0; ch < NCHUNKS; ++ch) {
        __syncthreads();
        // cooperative load of 64-code chunk into padded LDS
        const float* cbsrc = cb + (long)ch * CHUNK * DDIM;
        #pragma unroll
        for (int i = 0; i < 8; ++i) {
            const int e  = (i * 256 + tid) * 2;      // 4096 floats total
            const v2f d2 = *(const v2f*)(cbsrc + e);
            const int rr = e / DDIM, dd = e % DDIM;
            *(v2f*)(lds_cb + rr * CBS + dd) = d2;
        }
        __syncthreads();

        #pragma unroll
        for (int ct = 0; ct < 4; ++ct) {
            v8f acc = {0.f, 0.f, 0.f, 0.f, 0.f, 0.f, 0.f, 0.f};
            const float* cbase = lds_cb + (ct * 16 + lx) * CBS + 2 * half;
            #pragma unroll
            for (int j = 0; j < 16; ++j) {
                const v2f b = *(const v2f*)(cbase + 4 * j);
                acc = __builtin_amdgcn_wmma_f32_16x16x4_f32(
                        false, a[j], false, b, (short)0, acc, false, false);
            }
            const int   n   = ch * CHUNK + ct * 16 + lx;
            const float c2n = c2[n];
            #pragma unroll
            for (int r = 0; r < 8; ++r) {
                const float s = c2n - 2.0f * acc[r];
                if (s < best[r]) { best[r] = s; bidx[r] = n; }
            }
        }
    }

    // ---- reduce min/argmin across the 16 lanes of each half (first-min ties)
    #pragma unroll
    for (int r = 0; r < 8; ++r) {
        float bv = best[r];
        int   bi = bidx[r];
        #pragma unroll
        for (int m = 8; m >= 1; m >>= 1) {
            const float ov = __shfl_xor(bv, m);
            const int   oi = __shfl_xor(bi, m);
            if (ov < bv || (ov == bv && oi < bi)) { bv = ov; bi = oi; }
        }
        best[r] = bv + v2m[r];   // errs2 for token m = r + 8*half
        bidx[r] = bi;
    }

    // ---- scalar outputs: errs2 and shortcodes
    if (lx == 0) {
        #pragma unroll
        for (int r = 0; r < 8; ++r) {
            const long t = tok0 + r + 8 * half;
            out_e[t] = best[r];
            out_z[t] = (float)bidx[r];
        }
    }

    // ---- vecs_hat = c[z] gather, coalesced 128B stores per row half
    #pragma unroll
    for (int m = 0; m < 16; ++m) {
        const int  zm = __shfl(bidx[m & 7], (m >> 3) << 4);
        const long t  = tok0 + m;
        out_hat[t * DDIM + lane]      = cb[zm * DDIM + lane];
        out_hat[t * DDIM + lane + 32] = cb[zm * DDIM + lane + 32];
    }

    // ---- deterministic per-block loss partial
    float lsum = 0.0f;
    if (lx == 0) {
        const long hs = (long)H * S;
        #pragma unroll
        for (int r = 0; r < 8; ++r) {
            const long t = tok0 + r + 8 * half;
            const long b = t / hs;
            const long s = t % S;
            lsum += mask[b * S + s] * best[r];
        }
    }
    lsum += __shfl_xor(lsum, 16);
    if (lane == 0) lds_part[wave] = lsum;
    __syncthreads();
    if (tid == 0) {
        float accp = 0.0f;
        for (int w = 0; w < WAVES; ++w) accp += lds_part[w];
        partials[blockIdx.x] = accp;
    }
}

// ---------------------------------------------------------------------------
// Kernel 3: deterministic final reduction of per-block partials
// ---------------------------------------------------------------------------
__global__ __launch_bounds__(256) void vq_finish(const float* __restrict__ partials,
                                                 int n, float* __restrict__ out_commit,
                                                 float inv_bs) {
    __shared__ float sh[256];
    float a = 0.0f;
    for (int i = threadIdx.x; i < n; i += 256) a += partials[i];
    sh[threadIdx.x] = a;
    __syncthreads();
    for (int s = 128; s > 0; s >>= 1) {
        if (threadIdx.x < s) sh[threadIdx.x] += sh[threadIdx.x + s];
        __syncthreads();
    }
    if (threadIdx.x == 0) *out_commit = sh[0] * inv_bs;
}

// ---------------------------------------------------------------------------
extern "C" void kernel_launch(void* const* d_in, const int* in_sizes, int n_in,
                              void* d_out, int out_size, void* d_ws, size_t ws_size,
                              hipStream_t stream) {
    const float* vecs = (const float*)d_in[0];   // [B,H,S,64] f32
    const float* mask = (const float*)d_in[1];   // [B,S] f32

    const long NV     = (long)in_sizes[0];       // B*H*S*64
    const long NM     = (long)in_sizes[1];       // B*S
    const long tokens = NV / DDIM;               // B*H*S
    const int  H      = (int)(NV / (DDIM * NM)); // 8
    const int  S      = 4096;                    // per reference setup

    // workspace layout
    float* cb       = (float*)d_ws;                       // 512*64 floats
    float* c2       = cb + LCODES * DDIM;                 // 512 floats
    float* partials = c2 + LCODES;                        // nblocks floats

    // output layout (flat concat, all f32)
    float* out_hat    = (float*)d_out;
    float* out_z      = out_hat + NV;
    float* out_e      = out_z + tokens;
    float* out_commit = out_e + tokens;

    const int nblocks = (int)(tokens / (WAVES * TILE));   // 1024

    vq_codebook<<<LCODES, 64, 0, stream>>>(cb, c2);
    vq_main<<<nblocks, 256, 0, stream>>>(vecs, mask, cb, c2,
                                         out_hat, out_z, out_e, partials, H, S);
    vq_finish<<<1, 256, 0, stream>>>(partials, nblocks, out_commit,
                                     1.0f / (float)NM);
}